// NodeGraph_88854283420044
// MI455X (gfx1250) — compile-verified
//
#include <hip/hip_runtime.h>
#include <math.h>

#define N_NODES 2048
#define D       128
#define A_MAX   256
#define K_CONN  32
#define ALPHA_F 0.98f

typedef float v2f __attribute__((ext_vector_type(2)));
typedef float v8f __attribute__((ext_vector_type(8)));

#if __has_builtin(__builtin_amdgcn_wmma_f32_16x16x4_f32)
#define HAVE_WMMA_F32 1
#else
#define HAVE_WMMA_F32 0
#pragma message("NOTE: __builtin_amdgcn_wmma_f32_16x16x4_f32 unavailable -> scalar fallback path")
#endif

#if __has_builtin(__builtin_amdgcn_global_load_async_to_lds_b128) && \
    __has_builtin(__builtin_amdgcn_s_wait_asynccnt)
#define HAVE_ASYNC_LDS 1
typedef int v4i __attribute__((ext_vector_type(4)));
typedef __attribute__((address_space(1))) v4i* glb_v4i_p;  // global (AS1)
typedef __attribute__((address_space(3))) v4i* lds_v4i_p;  // LDS (AS3)
#else
#define HAVE_ASYNC_LDS 0
#pragma message("NOTE: global_load_async_to_lds builtins unavailable -> plain LDS staging")
#endif

// ---------------- wave32 helpers (no dependence on wavefront-size macros) ----
__device__ __forceinline__ float lane_xor_f32(float x, int m) {
  const int lane = (int)(threadIdx.x & 31u);
  return __int_as_float(
      __builtin_amdgcn_ds_bpermute(((lane ^ m) << 2), __float_as_int(x)));
}
__device__ __forceinline__ float wave_sum(float x) {
#pragma unroll
  for (int m = 16; m >= 1; m >>= 1) x += lane_xor_f32(x, m);
  return x;
}

// ---------------- WMMA f32 GEMV: out(128) = M(128x128) * v  (or M^T * v) ----
// One full wave executes this (EXEC all ones; no divergence around WMMA).
// A tile layout (16x4 f32): lanes 0-15 -> M rows, VGPR0/1 = K {0,1} (lo half)
// or K {2,3} (hi half). B (4x16) is v replicated across all 16 columns, so
// every column of C equals the GEMV tile; lanes with N==0 extract it.
template <bool TRANS>
__device__ __forceinline__ void wmma_gemv_128(const float* __restrict__ M,
                                              const float* __restrict__ vsm,
                                              float* __restrict__ out) {
  const int  lane = (int)(threadIdx.x & 31u);
  const bool hi   = lane >= 16;
  const int  m    = lane & 15;
#pragma unroll 1
  for (int tile = 0; tile < 8; ++tile) {
#if HAVE_WMMA_F32
    v8f c = {0.f, 0.f, 0.f, 0.f, 0.f, 0.f, 0.f, 0.f};
#else
    float s = 0.f;
#endif
#pragma unroll 8
    for (int kc = 0; kc < 32; ++kc) {
      const int k0 = kc * 4 + (hi ? 2 : 0);
      v2f a, b;
      if (TRANS) {
        a.x = M[(size_t)(k0)*D + tile * 16 + m];
        a.y = M[(size_t)(k0 + 1) * D + tile * 16 + m];
      } else {
        const float2 t =
            *reinterpret_cast<const float2*>(&M[(size_t)(tile * 16 + m) * D + k0]);
        a.x = t.x;
        a.y = t.y;
      }
      b.x = vsm[k0];
      b.y = vsm[k0 + 1];
#if HAVE_WMMA_F32
      c = __builtin_amdgcn_wmma_f32_16x16x4_f32(false, a, false, b, (short)0, c,
                                                false, false);
#else
      s += a.x * b.x + a.y * b.y;
#endif
    }
#if HAVE_WMMA_F32
    if (m == 0) {  // lane 0 -> rows tile*16+0..7, lane 16 -> rows tile*16+8..15
      const int rbase = tile * 16 + (hi ? 8 : 0);
#pragma unroll
      for (int j = 0; j < 8; ++j) out[rbase + j] = c[j];
    }
#else
    const float tot = s + lane_xor_f32(s, 16);
    if (!hi) out[tile * 16 + m] = tot;
#endif
  }
}

// ---------------- K1: activation gating --------------------------------------
__global__ void k_act_gate(const float* __restrict__ state,
                           const float* __restrict__ act_w,
                           const float* __restrict__ rand_u,
                           int* __restrict__ mask) {
  const int wid  = (int)(threadIdx.x >> 5);
  const int lane = (int)(threadIdx.x & 31u);
  const int n    = (int)blockIdx.x * 8 + wid;
  const float4 s = *(const float4*)&state[(size_t)n * D + lane * 4];
  const float4 w = *(const float4*)&act_w[(size_t)n * D + lane * 4];
  float d = s.x * w.x + s.y * w.y + s.z * w.z + s.w * w.w;
  d = wave_sum(d);
  if (lane == 0) {
    const float sig = 1.f / (1.f + expf(-d));
    mask[n] = (sig > rand_u[n]) ? 1 : 0;
  }
}

// ---------------- K2: ordered compaction (single wave) -----------------------
__global__ void k_compact(const int* __restrict__ mask, int* __restrict__ act_raw,
                          int* __restrict__ is_zeroed) {
  const int lane = (int)threadIdx.x;  // 0..31
  int base = 0;
  for (int c = 0; c < N_NODES / 32; ++c) {
    const int n = c * 32 + lane;
    const int m = mask[n];
    const unsigned b = __builtin_amdgcn_ballot_w32(m != 0);
    const unsigned lmask = (lane == 0) ? 0u : ((1u << lane) - 1u);
    const int pos = base + (int)__popc(b & lmask);
    if (m && pos < A_MAX) {
      act_raw[pos] = n;
      is_zeroed[n] = 1;  // ns rows zeroed only for the first A_MAX actives
    }
    base += (int)__popc(b);
  }
  for (int i = base + lane; i < A_MAX; i += 32) act_raw[i] = -1;
  // null slots clamp -1 -> 0, so reference zeroes node_state row 0
  if (lane == 0 && base < A_MAX) is_zeroed[0] = 1;
}

// ---------------- K3: act_vals[a] = nodes[e] @ state[e] ----------------------
__global__ void k_act_vals(const int* __restrict__ act_raw,
                           const float* __restrict__ state,
                           const float* __restrict__ nodes,
                           float* __restrict__ act_vals) {
  const int a = (int)blockIdx.x;
  const int e = act_raw[a];
  if (e < 0) return;  // null slot: contributes nothing downstream
  __shared__ float vsm[D];
  const int lane = (int)threadIdx.x;  // 0..31
#if HAVE_ASYNC_LDS
  __builtin_amdgcn_global_load_async_to_lds_b128(
      (glb_v4i_p)&state[(size_t)e * D + lane * 4], (lds_v4i_p)&vsm[lane * 4], 0, 0);
  __builtin_amdgcn_s_wait_asynccnt(0);
#else
  *(float4*)&vsm[lane * 4] = *(const float4*)&state[(size_t)e * D + lane * 4];
#endif
  __syncthreads();
  wmma_gemv_128<false>(nodes + (size_t)e * D * D, vsm, act_vals + (size_t)a * D);
}

// ---------------- K4: softmax + top-32 (softmax is monotone -> topk on raw) --
__global__ void k_softmax_topk(const int* __restrict__ act_raw,
                               const float* __restrict__ connections,
                               int* __restrict__ top_i, float* __restrict__ top_w) {
  const int a = (int)blockIdx.x;
  const int e = act_raw[a];
  if (e < 0) return;
  __shared__ float sv[N_NODES];
  __shared__ float rv[256];
  __shared__ int   ri[256];
  __shared__ float s_max, s_inv;
  const int t = (int)threadIdx.x;
#if HAVE_ASYNC_LDS
#pragma unroll
  for (int j = t * 4; j < N_NODES; j += 256 * 4)
    __builtin_amdgcn_global_load_async_to_lds_b128(
        (glb_v4i_p)&connections[(size_t)e * N_NODES + j], (lds_v4i_p)&sv[j], 0, 0);
  __builtin_amdgcn_s_wait_asynccnt(0);
#else
  for (int j = t; j < N_NODES; j += 256)
    sv[j] = connections[(size_t)e * N_NODES + j];
#endif
  __syncthreads();

  float mx = -INFINITY;
#pragma unroll
  for (int i = 0; i < 8; ++i) mx = fmaxf(mx, sv[t * 8 + i]);
  rv[t] = mx;
  __syncthreads();
  for (int s = 128; s > 0; s >>= 1) {
    if (t < s) rv[t] = fmaxf(rv[t], rv[t + s]);
    __syncthreads();
  }
  if (t == 0) s_max = rv[0];
  __syncthreads();
  const float rmax = s_max;

  float acc = 0.f;
#pragma unroll
  for (int i = 0; i < 8; ++i) acc += expf(sv[t * 8 + i] - rmax);
  rv[t] = acc;
  __syncthreads();
  for (int s = 128; s > 0; s >>= 1) {
    if (t < s) rv[t] += rv[t + s];
    __syncthreads();
  }
  if (t == 0) s_inv = 1.f / rv[0];
  __syncthreads();
  const float inv = s_inv;

  for (int k = 0; k < K_CONN; ++k) {
    float bv = -INFINITY;
    int   bi = 0x7fffffff;
#pragma unroll
    for (int i = 0; i < 8; ++i) {
      const float v = sv[t * 8 + i];
      if (v > bv) { bv = v; bi = t * 8 + i; }  // first (lowest idx) on ties
    }
    rv[t] = bv; ri[t] = bi;
    __syncthreads();
    for (int s = 128; s > 0; s >>= 1) {
      if (t < s) {
        if (rv[t + s] > rv[t] || (rv[t + s] == rv[t] && ri[t + s] < ri[t])) {
          rv[t] = rv[t + s];
          ri[t] = ri[t + s];
        }
      }
      __syncthreads();
    }
    if (t == 0) {
      const int idx = ri[0];
      top_i[a * K_CONN + k] = idx;
      top_w[a * K_CONN + k] = expf(rv[0] - rmax) * inv;
      sv[idx] = -INFINITY;  // mask out for next iteration
    }
    __syncthreads();
  }
}

// ---------------- K4b: scatter-add propagation -------------------------------
__global__ void k_scatter(const int* __restrict__ act_raw,
                          const int* __restrict__ top_i,
                          const float* __restrict__ top_w,
                          const float* __restrict__ act_vals,
                          float* __restrict__ node_inputs) {
  const int a = (int)blockIdx.y;
  if (act_raw[a] < 0) return;
  const int k   = (int)blockIdx.x;
  const int tgt = top_i[a * K_CONN + k];
  const float w = top_w[a * K_CONN + k];
  const int d   = (int)threadIdx.x;
  atomicAdd(&node_inputs[(size_t)tgt * D + d], w * act_vals[(size_t)a * D + d]);
}

// ---------------- K5: next_pred (WMMA M^T v), loss, epilogue -----------------
__global__ void k_predict(const float* __restrict__ state,
                          const float* __restrict__ input_pred,
                          const float* __restrict__ pred_nodes,
                          const float* __restrict__ pred_scale,
                          const float* __restrict__ pred_offset,
                          const float* __restrict__ node_exh,
                          const int* __restrict__ is_zeroed,
                          const float* __restrict__ node_inputs,
                          float* __restrict__ out_state,
                          float* __restrict__ out_exh,
                          float* __restrict__ out_pred,
                          float* __restrict__ loss_acc) {
  __shared__ float sin_[8][D];
  __shared__ float snp[8][D];
  const int wid  = (int)(threadIdx.x >> 5);
  const int lane = (int)(threadIdx.x & 31u);
  const int n    = (int)blockIdx.x * 8 + wid;
  const int d0   = lane * 4;

  const float4 ni = *(const float4*)&node_inputs[(size_t)n * D + d0];
  *(float4*)&sin_[wid][d0] = ni;

  const float4 ip = *(const float4*)&input_pred[(size_t)n * D + d0];
  float l = 0.f;
  {
    float dx;
    dx = ip.x - ni.x; l += dx * dx;
    dx = ip.y - ni.y; l += dx * dx;
    dx = ip.z - ni.z; l += dx * dx;
    dx = ip.w - ni.w; l += dx * dx;
  }
  l *= 0.5f;

  __syncthreads();
  wmma_gemv_128<true>(pred_nodes + (size_t)n * D * D, sin_[wid], snp[wid]);
  __syncthreads();

  const float4 np = *(const float4*)&snp[wid][d0];
  const float4 sc = *(const float4*)&pred_scale[(size_t)n * D + d0];
  const float4 of = *(const float4*)&pred_offset[(size_t)n * D + d0];
  float4 st = make_float4(0.f, 0.f, 0.f, 0.f);
  if (!is_zeroed[n]) st = *(const float4*)&state[(size_t)n * D + d0];
  float4 o;
  o.x = (st.x + ni.x + np.x * sc.x + of.x) * ALPHA_F;
  o.y = (st.y + ni.y + np.y * sc.y + of.y) * ALPHA_F;
  o.z = (st.z + ni.z + np.z * sc.z + of.z) * ALPHA_F;
  o.w = (st.w + ni.w + np.w * sc.w + of.w) * ALPHA_F;
  *(float4*)&out_state[(size_t)n * D + d0] = o;
  *(float4*)&out_pred[(size_t)n * D + d0]  = np;

  l = wave_sum(l);
  if (lane == 0) {
    out_exh[n] = node_exh[n];
    atomicAdd(loss_acc, l);
  }
}

__global__ void k_finalize(const float* __restrict__ loss_acc,
                           float* __restrict__ out_loss) {
  *out_loss = *loss_acc * (1.f / (float)(N_NODES * D));
}

// ---------------- host launcher ---------------------------------------------
extern "C" void kernel_launch(void* const* d_in, const int* in_sizes, int n_in,
                              void* d_out, int out_size, void* d_ws,
                              size_t ws_size, hipStream_t stream) {
  (void)in_sizes; (void)n_in; (void)out_size; (void)ws_size;
  const float* node_state  = (const float*)d_in[0];
  const float* node_exh    = (const float*)d_in[1];
  const float* input_pred  = (const float*)d_in[2];
  const float* rand_u      = (const float*)d_in[3];
  const float* nodes       = (const float*)d_in[4];
  const float* pred_nodes  = (const float*)d_in[5];
  const float* nodes_act   = (const float*)d_in[6];
  const float* pred_scale  = (const float*)d_in[7];
  const float* pred_offset = (const float*)d_in[8];
  const float* connections = (const float*)d_in[9];

  // workspace layout (4-byte units)
  char* ws = (char*)d_ws;
  int*   mask        = (int*)(ws + (size_t)0 * 4);       // 2048
  int*   act_raw     = (int*)(ws + (size_t)2048 * 4);    // 256
  float* act_vals    = (float*)(ws + (size_t)2320 * 4);  // 256*128
  int*   top_i       = (int*)(ws + (size_t)35088 * 4);   // 256*32
  float* top_w       = (float*)(ws + (size_t)43280 * 4); // 256*32
  int*   is_zeroed   = (int*)(ws + (size_t)51472 * 4);   // 2048   } zero-init
  float* node_inputs = (float*)(ws + (size_t)53520 * 4); // 262144 } (one
  float* loss_acc    = (float*)(ws + (size_t)315664 * 4);// 1      }  memset)

  float* out_state = (float*)d_out;                    // 2048*128
  float* out_exh   = out_state + (size_t)N_NODES * D;  // 2048
  float* out_pred  = out_exh + N_NODES;                // 2048*128
  float* out_loss  = out_pred + (size_t)N_NODES * D;   // 1

  (void)hipMemsetAsync(is_zeroed, 0, (size_t)(2048 + N_NODES * D + 1) * 4, stream);

  k_act_gate<<<N_NODES / 8, 256, 0, stream>>>(node_state, nodes_act, rand_u, mask);
  k_compact<<<1, 32, 0, stream>>>(mask, act_raw, is_zeroed);
  k_act_vals<<<A_MAX, 32, 0, stream>>>(act_raw, node_state, nodes, act_vals);
  k_softmax_topk<<<A_MAX, 256, 0, stream>>>(act_raw, connections, top_i, top_w);
  dim3 sg(K_CONN, A_MAX);
  k_scatter<<<sg, D, 0, stream>>>(act_raw, top_i, top_w, act_vals, node_inputs);
  k_predict<<<N_NODES / 8, 256, 0, stream>>>(node_state, input_pred, pred_nodes,
                                             pred_scale, pred_offset, node_exh,
                                             is_zeroed, node_inputs, out_state,
                                             out_exh, out_pred, loss_acc);
  k_finalize<<<1, 1, 0, stream>>>(loss_acc, out_loss);
}